// DAGConstraintLayer_82970178224202
// MI455X (gfx1250) — compile-verified
//
#include <hip/hip_runtime.h>
#include <hip/hip_bf16.h>

// DAGConstraintLayer: out = tree-min-propagate(sigmoid(x)), x: [262144, 127] f32.
// Bandwidth-bound: ~266 MB traffic -> ~11.4 us floor at 23.3 TB/s on MI455X.
// No matmul structure -> WMMA inapplicable; use async global->LDS load path
// (CDNA5 async-to-LDS ops, ASYNCcnt) for the LDS staging we need anyway.

#define NUM_NODES 127
#define ROWS_PER_BLOCK 4
#define BLOCK_THREADS (ROWS_PER_BLOCK * 128)   // 16 waves (wave32)

#if defined(__has_builtin)
#  if __has_builtin(__builtin_amdgcn_global_load_async_to_lds_b32)
#    define HAVE_ASYNC_LDS 1
#  else
#    define HAVE_ASYNC_LDS 0
#  endif
#else
#  define HAVE_ASYNC_LDS 0
#endif

typedef __attribute__((address_space(1))) int g_int;
typedef __attribute__((address_space(3))) int l_int;

__device__ __forceinline__ void wait_async_all() {
#if defined(__has_builtin) && __has_builtin(__builtin_amdgcn_s_wait_asynccnt)
    __builtin_amdgcn_s_wait_asynccnt(0);
#else
    asm volatile("s_wait_asynccnt 0" ::: "memory");
#endif
}

__global__ __launch_bounds__(BLOCK_THREADS)
void DAGConstraintLayer_82970178224202_kernel(const float* __restrict__ x,
                                              float* __restrict__ out,
                                              int nrows) {
    // one padded 128-float slot per row; waves never straddle rows
    __shared__ float s[ROWS_PER_BLOCK * 128];

    const int tid  = threadIdx.x;
    const int rl   = tid >> 7;        // row within block
    const int node = tid & 127;       // tree node (127 valid, lane 127 idles)
    const int row  = blockIdx.x * ROWS_PER_BLOCK + rl;
    const int base = rl * 128;
    const bool active = (node < NUM_NODES) && (row < nrows);
    const long long g = (long long)row * NUM_NODES + node;

#if HAVE_ASYNC_LDS
    // CDNA5 async copy: global -> LDS without touching VGPR data path.
    if (active) {
        __builtin_amdgcn_global_load_async_to_lds_b32(
            (g_int*)(x + g),
            (l_int*)(&s[base + node]),
            /*offset=*/0, /*cpol=*/0);
    }
    wait_async_all();      // this wave's async writes to LDS are done
    __syncthreads();       // all waves' writes visible

    float v = 0.0f;
    if (active) {
        float raw = s[base + node];
        v = 1.0f / (1.0f + __expf(-raw));
        s[base + node] = v;            // overwrite raw with sigmoid
    }
    __syncthreads();
#else
    float v = 0.0f;
    if (active) {
        float raw = x[g];              // coalesced 4B/lane
        v = 1.0f / (1.0f + __expf(-raw));
        s[base + node] = v;
    }
    __syncthreads();
#endif

    if (active) {
        // min over root-to-node path: at most 6 ancestors above any node
        float m = v;
        int cur = node;
#pragma unroll
        for (int k = 0; k < 6; ++k) {
            if (cur > 0) {
                cur = (cur - 1) >> 1;
                m = fminf(m, s[base + cur]);   // LDS broadcast-friendly
            }
        }
        out[g] = m;                    // coalesced store
    }
}

extern "C" void kernel_launch(void* const* d_in, const int* in_sizes, int n_in,
                              void* d_out, int out_size, void* d_ws, size_t ws_size,
                              hipStream_t stream) {
    const float* x = (const float*)d_in[0];
    float* out = (float*)d_out;
    const int nrows = in_sizes[0] / NUM_NODES;          // 262144
    const int grid  = (nrows + ROWS_PER_BLOCK - 1) / ROWS_PER_BLOCK;
    DAGConstraintLayer_82970178224202_kernel<<<grid, BLOCK_THREADS, 0, stream>>>(x, out, nrows);
}